// SparseRNN_54082228191947
// MI455X (gfx1250) — compile-verified
//
#include <hip/hip_runtime.h>

// SpMM scatter:  out[row,:] += val * inp[col,:]   (COO, fp32, D = 128)
// Phase 1: out[n,d] = bias[n]
// Phase 2: per-wave edge scatter with native no-return f32 atomics.
//
// Roofline (MI455X): 1.66 GFLOP vs ~180 MB HBM traffic (inp/out are 51.2 MB
// each -> L2-resident in 192 MB L2). Limiter = 832M fp32 atomic-adds into L2.
// So: coalesced metadata streams, B128 row gathers, wave-uniform broadcast via
// v_readlane, and relaxed agent-scope no-return global_atomic_add_f32.

#define D_FEAT 128          // feature width (reference: D = 128)
#define D_SHIFT 7           // log2(D_FEAT)

// ---------------------------------------------------------------- init kernel
__global__ void spmm_init_bias(const float* __restrict__ bias,
                               float* __restrict__ out,
                               int total4 /* = N*D/4 */) {
    int i = blockIdx.x * blockDim.x + threadIdx.x;
    if (i < total4) {
        // 4 floats per thread; row index = (element index) / 128
        int row = (i << 2) >> D_SHIFT;
        float b = bias[row];
        float4 v = make_float4(b, b, b, b);
        reinterpret_cast<float4*>(out)[i] = v;
    }
}

// ------------------------------------------------------------- scatter kernel
// One wave (32 lanes) processes 32 edges. Metadata loads are fully coalesced.
// For each edge, the wave gathers one full 512B input row (float4 per lane)
// and scatter-adds it into the destination row with global_atomic_add_f32.
__global__ void spmm_scatter(const float* __restrict__ inp,
                             const int*   __restrict__ rows,
                             const int*   __restrict__ cols,
                             const float* __restrict__ vals,
                             float*       __restrict__ out,
                             int nnz) {
    const int lane = threadIdx.x & 31;
    const int wave = (blockIdx.x * blockDim.x + threadIdx.x) >> 5;
    const int e0   = wave << 5;                 // first edge of this wave
    if (e0 >= nnz) return;

    const int e = e0 + lane;
    int   r = 0, c = 0;
    int   vbits = 0;
    if (e < nnz) {
        r = rows[e];
        c = cols[e];
        vbits = __float_as_int(vals[e]);
    }

    // Warm L0/L2 for this wave's 32 gather rows: each lane prefetches the
    // 4 cachelines (512B) of its own edge's source row. gfx1250-specific
    // global_prefetch_b8 path.
    {
        const char* p = reinterpret_cast<const char*>(inp + ((size_t)c << D_SHIFT));
        __builtin_prefetch(p,       0, 1);
        __builtin_prefetch(p + 128, 0, 1);
        __builtin_prefetch(p + 256, 0, 1);
        __builtin_prefetch(p + 384, 0, 1);
    }

    const int cnt = min(32, nnz - e0);
    const int fo  = lane << 2;                  // this lane's feature offset

    for (int k = 0; k < cnt; ++k) {
        // wave-uniform broadcast of edge k's metadata: v_readlane_b32 -> SGPR
        const int   rk = __builtin_amdgcn_readlane(r, k);
        const int   ck = __builtin_amdgcn_readlane(c, k);
        const float vk = __int_as_float(__builtin_amdgcn_readlane(vbits, k));

        // coalesced 512B row gather: global_load_b128 (SGPR base + lane offset)
        const float4 x = *reinterpret_cast<const float4*>(
            inp + ((size_t)ck << D_SHIFT) + fo);

        float* o = out + ((size_t)rk << D_SHIFT) + fo;

        // native fp32 atomic adds, relaxed / agent scope, result discarded
        // -> no-return global_atomic_add_f32 hitting L2 atomic units
        (void)__hip_atomic_fetch_add(o + 0, vk * x.x,
                                     __ATOMIC_RELAXED, __HIP_MEMORY_SCOPE_AGENT);
        (void)__hip_atomic_fetch_add(o + 1, vk * x.y,
                                     __ATOMIC_RELAXED, __HIP_MEMORY_SCOPE_AGENT);
        (void)__hip_atomic_fetch_add(o + 2, vk * x.z,
                                     __ATOMIC_RELAXED, __HIP_MEMORY_SCOPE_AGENT);
        (void)__hip_atomic_fetch_add(o + 3, vk * x.w,
                                     __ATOMIC_RELAXED, __HIP_MEMORY_SCOPE_AGENT);
    }
}

// ------------------------------------------------------------------- launcher
extern "C" void kernel_launch(void* const* d_in, const int* in_sizes, int n_in,
                              void* d_out, int out_size, void* d_ws, size_t ws_size,
                              hipStream_t stream) {
    // setup_inputs() order: inp, rows, cols, vals, bias
    const float* inp  = (const float*)d_in[0];
    const int*   rows = (const int*)  d_in[1];
    const int*   cols = (const int*)  d_in[2];
    const float* vals = (const float*)d_in[3];
    const float* bias = (const float*)d_in[4];
    float*       out  = (float*)d_out;

    const int nnz    = in_sizes[1];      // length of rows == cols == vals
    const int total  = out_size;         // N * D
    const int total4 = total >> 2;

    // Phase 1: out = bias (broadcast over the feature dimension)
    {
        int threads = 256;
        int blocks  = (total4 + threads - 1) / threads;
        spmm_init_bias<<<blocks, threads, 0, stream>>>(bias, out, total4);
    }

    // Phase 2: edge scatter. One wave per 32 edges; 8 waves (256 threads)
    // per block -> 256 edges per block.
    {
        int waves   = (nnz + 31) / 32;
        int blocks  = (waves + 7) / 8;
        spmm_scatter<<<blocks, 256, 0, stream>>>(inp, rows, cols, vals, out, nnz);
    }
}